// AC_loss_79998060855359
// MI455X (gfx1250) — compile-verified
//
#include <hip/hip_runtime.h>
#include <math.h>

typedef __attribute__((ext_vector_type(2))) float v2f;
typedef __attribute__((ext_vector_type(8))) float v8f;

#define B_  16
#define C_  64
#define H_  256
#define W_  256
#define HW_ (H_ * W_)

// ---------------------------------------------------------------------------
// Kernel 1: fused streaming pass. One block per (b,c) plane.
// Computes argmax(predict plane), argmax(gt plane), and sum((p-g)^2) in a
// single read of each tensor (537 MB total -> ~23 us HBM floor @ 23.3 TB/s).
// float4 loads -> global_load_b128; prefetch hint -> global_prefetch_b8.
// ---------------------------------------------------------------------------
__global__ __launch_bounds__(256) void k1_argmax_sq(
    const float* __restrict__ p, const float* __restrict__ g,
    unsigned* __restrict__ p_idx, unsigned* __restrict__ g_idx,
    float* __restrict__ sq) {
  __shared__ float    s_pv[256]; __shared__ unsigned s_pi[256];
  __shared__ float    s_gv[256]; __shared__ unsigned s_gi[256];
  __shared__ float    s_ss[256];

  const int bc  = blockIdx.x;
  const int tid = threadIdx.x;
  const size_t base = (size_t)bc * HW_;
  const float4* p4 = (const float4*)(p + base);
  const float4* g4 = (const float4*)(g + base);

  float pbest = -INFINITY; unsigned pbi = 0;
  float gbest = -INFINITY; unsigned gbi = 0;
  float ss = 0.0f;

  const int iters = HW_ / 4 / 256;   // 64
  for (int k = 0; k < iters; ++k) {
    const int f = k * 256 + tid;
    if (k + 1 < iters) {             // speculative prefetch of next chunk
      __builtin_prefetch(&p4[f + 256], 0, 0);
      __builtin_prefetch(&g4[f + 256], 0, 0);
    }
    float4 pv = p4[f];
    float4 gv = g4[f];
    const unsigned e = (unsigned)f * 4u;
    float pc[4] = {pv.x, pv.y, pv.z, pv.w};
    float gc[4] = {gv.x, gv.y, gv.z, gv.w};
#pragma unroll
    for (int c = 0; c < 4; ++c) {
      if (pc[c] > pbest) { pbest = pc[c]; pbi = e + c; }
      if (gc[c] > gbest) { gbest = gc[c]; gbi = e + c; }
      const float d = pc[c] - gc[c];
      ss += d * d;
    }
  }

  s_pv[tid] = pbest; s_pi[tid] = pbi;
  s_gv[tid] = gbest; s_gi[tid] = gbi;
  s_ss[tid] = ss;
  __syncthreads();
#pragma unroll
  for (int s = 128; s > 0; s >>= 1) {
    if (tid < s) {
      if (s_pv[tid + s] > s_pv[tid] ||
          (s_pv[tid + s] == s_pv[tid] && s_pi[tid + s] < s_pi[tid])) {
        s_pv[tid] = s_pv[tid + s]; s_pi[tid] = s_pi[tid + s];
      }
      if (s_gv[tid + s] > s_gv[tid] ||
          (s_gv[tid + s] == s_gv[tid] && s_gi[tid + s] < s_gi[tid])) {
        s_gv[tid] = s_gv[tid + s]; s_gi[tid] = s_gi[tid + s];
      }
      s_ss[tid] += s_ss[tid + s];
    }
    __syncthreads();
  }
  if (tid == 0) {
    p_idx[bc] = s_pi[0];
    g_idx[bc] = s_gi[0];
    sq[bc]    = s_ss[0];
  }
}

// ---------------------------------------------------------------------------
// Kernel 2: per-batch WMMA epilogue. One 128-thread block (4 waves) per batch.
// Gram matrix G = XY * XY^T (XY in [64,2]) via V_WMMA_F32_16X16X4_F32 with
// K padded 2 -> 4 (zeros). Wave ti handles tile-row ti; each lane then
// post-processes its 8 C/D elements: dist = sqrt(ni + nj - 2G), angle =
// acos(clip(G / max(|xi||xj|, 1e-8))), accumulating both MSEs.
// ---------------------------------------------------------------------------
__global__ __launch_bounds__(128) void k2_wmma_loss(
    const unsigned* __restrict__ p_idx, const unsigned* __restrict__ g_idx,
    const float* __restrict__ sq, float* __restrict__ partial) {
  __shared__ float s_p[C_][2];
  __shared__ float s_g[C_][2];
  __shared__ float s_pn[C_];
  __shared__ float s_gn[C_];
  __shared__ float r_d[128];
  __shared__ float r_a[128];
  __shared__ float r_l[128];

  const int b   = blockIdx.x;
  const int tid = threadIdx.x;

  // Decode argmax indices -> (row, col) float coords + squared norms.
  if (tid < C_) {
    const unsigned ip = p_idx[b * C_ + tid];
    const float r = (float)(ip / W_), c = (float)(ip % W_);
    s_p[tid][0] = r; s_p[tid][1] = c; s_pn[tid] = r * r + c * c;
  } else if (tid < 2 * C_) {
    const int cc = tid - C_;
    const unsigned ig = g_idx[b * C_ + cc];
    const float r = (float)(ig / W_), c = (float)(ig % W_);
    s_g[cc][0] = r; s_g[cc][1] = c; s_gn[cc] = r * r + c * c;
  }
  __syncthreads();

  const int ti   = tid >> 5;      // wave id = tile row
  const int lane = tid & 31;
  const int half = lane >> 4;     // lanes 16-31 hold K=2,3 (zero padding)
  const int m    = lane & 15;
  const float sel = (half == 0) ? 1.0f : 0.0f;

  float acc_d = 0.0f, acc_a = 0.0f;

  for (int tj = 0; tj < 4; ++tj) {
    const int ri = ti * 16 + m;   // A row (M = m) for lanes 0-15
    const int rj = tj * 16 + m;   // B col (N = m) for lanes 0-15
    v2f ap, bp, ag, bg;
    // A layout (32-bit 16x4): VGPR0 = K0 (lanes 0-15) / K2 (16-31, zero);
    //                         VGPR1 = K1 / K3.  B is the mirrored transpose.
    ap[0] = s_p[ri][0] * sel; ap[1] = s_p[ri][1] * sel;
    bp[0] = s_p[rj][0] * sel; bp[1] = s_p[rj][1] * sel;
    ag[0] = s_g[ri][0] * sel; ag[1] = s_g[ri][1] * sel;
    bg[0] = s_g[rj][0] * sel; bg[1] = s_g[rj][1] * sel;

    v8f cz = {0.f, 0.f, 0.f, 0.f, 0.f, 0.f, 0.f, 0.f};
    v8f Gp = __builtin_amdgcn_wmma_f32_16x16x4_f32(
        false, ap, false, bp, (short)0, cz, false, false);
    v8f Gg = __builtin_amdgcn_wmma_f32_16x16x4_f32(
        false, ag, false, bg, (short)0, cz, false, false);

#pragma unroll
    for (int v = 0; v < 8; ++v) {
      const int i = ti * 16 + v + 8 * half;   // C/D layout: M = v + 8*(lane>=16)
      const int j = tj * 16 + m;              //             N = lane & 15
      const float gp = Gp[v];
      const float gg = Gg[v];
      // cdist via Gram: ||xi - xj||^2 = ni + nj - 2*G[i,j]
      const float dp = sqrtf(fmaxf(s_pn[i] + s_pn[j] - 2.0f * gp, 0.0f));
      const float dg = sqrtf(fmaxf(s_gn[i] + s_gn[j] - 2.0f * gg, 0.0f));
      const float dd = dp - dg;
      acc_d += dd * dd;
      // angle matrix: acos(clip(cos_sim))
      const float denp = fmaxf(sqrtf(s_pn[i]) * sqrtf(s_pn[j]), 1e-8f);
      const float deng = fmaxf(sqrtf(s_gn[i]) * sqrtf(s_gn[j]), 1e-8f);
      const float cp = fminf(fmaxf(gp / denp, -1.0f + 1e-7f), 1.0f - 1e-7f);
      const float cg = fminf(fmaxf(gg / deng, -1.0f + 1e-7f), 1.0f - 1e-7f);
      const float aa = acosf(cp) - acosf(cg);
      acc_a += aa * aa;
    }
  }

  // Fold in the per-channel L2 partials for this batch, then block-reduce.
  r_d[tid] = acc_d;
  r_a[tid] = acc_a;
  r_l[tid] = (tid < C_) ? sq[b * C_ + tid] : 0.0f;
  __syncthreads();
#pragma unroll
  for (int s = 64; s > 0; s >>= 1) {
    if (tid < s) {
      r_d[tid] += r_d[tid + s];
      r_a[tid] += r_a[tid + s];
      r_l[tid] += r_l[tid + s];
    }
    __syncthreads();
  }
  if (tid == 0) {
    const float mseD = r_d[0] * (1.0f / (float)(C_ * C_));
    const float mseA = r_a[0] * (1.0f / (float)(C_ * C_));
    const float l2   = r_l[0] * (1.0f / ((float)C_ * (float)HW_));
    partial[b] = (log2f(mseD) + log2f(mseA)) * l2;
  }
}

// ---------------------------------------------------------------------------
// Kernel 3: deterministic scalar finisher (16 values -> 1).
// ---------------------------------------------------------------------------
__global__ void k3_final(const float* __restrict__ partial,
                         float* __restrict__ out) {
  if (threadIdx.x == 0 && blockIdx.x == 0) {
    float s = 0.0f;
    for (int b = 0; b < B_; ++b) s += partial[b];
    out[0] = s * (1.0f / (float)B_);
  }
}

// ---------------------------------------------------------------------------
extern "C" void kernel_launch(void* const* d_in, const int* in_sizes, int n_in,
                              void* d_out, int out_size, void* d_ws,
                              size_t ws_size, hipStream_t stream) {
  const float* predict = (const float*)d_in[0];
  const float* gt      = (const float*)d_in[1];

  // workspace layout: [p_idx u32 x 1024][g_idx u32 x 1024][sq f32 x 1024][partial f32 x 16]
  unsigned* p_idx   = (unsigned*)d_ws;
  unsigned* g_idx   = p_idx + B_ * C_;
  float*    sq      = (float*)(g_idx + B_ * C_);
  float*    partial = sq + B_ * C_;

  k1_argmax_sq<<<B_ * C_, 256, 0, stream>>>(predict, gt, p_idx, g_idx, sq);
  k2_wmma_loss<<<B_, 128, 0, stream>>>(p_idx, g_idx, sq, partial);
  k3_final<<<1, 32, 0, stream>>>(partial, (float*)d_out);
}